// GraphTransformer_80032420594209
// MI455X (gfx1250) — compile-verified
//
#include <hip/hip_runtime.h>
#include <math.h>

#define NN 20000
#define EE 320000

typedef __attribute__((ext_vector_type(2))) float v2f;
typedef __attribute__((ext_vector_type(8))) float v8f;

// ---------------------------------------------------------------- helpers
__device__ __forceinline__ void atomicMaxF(float* addr, float val) {
  // signed-int max for non-negative floats, unsigned-int min for negative floats.
  // Mixed updates are consistent because positive-bit-patterns < 0x80000000 <= negative-bit-patterns (unsigned),
  // and signed compare puts negatives below positives.
  if (val >= 0.0f) atomicMax((int*)addr, __float_as_int(val));
  else             atomicMin((unsigned int*)addr, (unsigned int)__float_as_int(val));
}

// ---------------------------------------------------------------- concat [x | pe] -> h0 (row stride 136)
__global__ void concat_kernel(const float* __restrict__ x, const float* __restrict__ pe,
                              float* __restrict__ h0, int n) {
  int i = blockIdx.x * blockDim.x + threadIdx.x;
  int total = n * 136;
  if (i >= total) return;
  int r = i / 136, c = i - r * 136;
  h0[i] = (c < 128) ? x[r * 128 + c] : pe[r * 8 + (c - 128)];
}

// ---------------------------------------------------------------- WMMA fp32 GEMM: C[M,fout] = A[M,K] @ W[K,fout] + bias
// One wave computes a 16(M) x 64(N) strip with 4 f32 accumulators, K in steps of 4
// (V_WMMA_F32_16X16X4_F32). blockDim = (32, fout/64); gridDim.x = M/16.
__global__ void wmma_gemm_bias(const float* __restrict__ A, int lda,
                               const float* __restrict__ W,
                               const float* __restrict__ bias,
                               float* __restrict__ C, int ldc, int K) {
  const int lane  = threadIdx.x;          // 0..31
  const int m16   = lane & 15;            // row within tile (A/B lane row)
  const int koff  = (lane >> 4) << 1;     // 0 for lanes 0-15 (K=0,1), 2 for lanes 16-31 (K=2,3)
  const int tileN = threadIdx.y * 64;

  const float* Arow = A + (size_t)(blockIdx.x * 16 + m16) * lda;

  v8f acc0 = {}, acc1 = {}, acc2 = {}, acc3 = {};

  for (int k0 = 0; k0 < K; k0 += 4) {
    v2f a;
    a.x = Arow[k0 + koff];          // VGPR0: K = koff
    a.y = Arow[k0 + koff + 1];      // VGPR1: K = koff+1

    const float* Wp = W + (size_t)(k0 + koff) * ldc + tileN + m16;
    v2f b0, b1, b2, b3;
    b0.x = Wp[0];      b0.y = Wp[ldc];
    b1.x = Wp[16];     b1.y = Wp[ldc + 16];
    b2.x = Wp[32];     b2.y = Wp[ldc + 32];
    b3.x = Wp[48];     b3.y = Wp[ldc + 48];

    acc0 = __builtin_amdgcn_wmma_f32_16x16x4_f32(false, a, false, b0, (short)0, acc0, false, false);
    acc1 = __builtin_amdgcn_wmma_f32_16x16x4_f32(false, a, false, b1, (short)0, acc1, false, false);
    acc2 = __builtin_amdgcn_wmma_f32_16x16x4_f32(false, a, false, b2, (short)0, acc2, false, false);
    acc3 = __builtin_amdgcn_wmma_f32_16x16x4_f32(false, a, false, b3, (short)0, acc3, false, false);
  }

  // C/D layout: VGPR r -> M = (lane<16 ? r : r+8), N = lane & 15
  const int crow = blockIdx.x * 16 + (lane >> 4) * 8;
  const int ccol = tileN + m16;
  const float bv0 = bias[ccol], bv1 = bias[ccol + 16], bv2 = bias[ccol + 32], bv3 = bias[ccol + 48];
#pragma unroll
  for (int r = 0; r < 8; ++r) {
    float* Cp = C + (size_t)(crow + r) * ldc + ccol;
    Cp[0]  = acc0[r] + bv0;
    Cp[16] = acc1[r] + bv1;
    Cp[32] = acc2[r] + bv2;
    Cp[48] = acc3[r] + bv3;
  }
}

// ---------------------------------------------------------------- per-layer softmax state init
__global__ void init_m_den(float* __restrict__ m, float* __restrict__ den, int n) {
  int i = blockIdx.x * blockDim.x + threadIdx.x;
  if (i < n) { m[i] = -__builtin_inff(); den[i] = 0.0f; }
}

// ---------------------------------------------------------------- one wave per (edge, head): score + segment max
__global__ void edge_score_max(const float* __restrict__ q, const float* __restrict__ kk,
                               const int* __restrict__ src, const int* __restrict__ dst,
                               float* __restrict__ sc, float* __restrict__ m,
                               int pairs, int H, int d, float scale) {
  int widx = blockIdx.x * (blockDim.x >> 5) + (threadIdx.x >> 5);
  if (widx >= pairs) return;
  int lane = threadIdx.x & 31;
  int e = widx / H, h = widx - e * H;
  int s = src[e], t = dst[e];
  const float* qp = q  + (size_t)t * H * d + h * d;   // q gathered by dst
  const float* kp = kk + (size_t)s * H * d + h * d;   // k gathered by src
  float acc = 0.0f;
  for (int i = lane; i < d; i += 32) acc += qp[i] * kp[i];
#pragma unroll
  for (int off = 16; off > 0; off >>= 1) acc += __shfl_xor(acc, off, 32);
  if (lane == 0) {
    float scv = acc * scale;
    sc[widx] = scv;
    atomicMaxF(m + (size_t)t * H + h, scv);
  }
}

// ---------------------------------------------------------------- exp(score - max) + segment sum
__global__ void edge_exp_sum(float* __restrict__ sc, const float* __restrict__ m,
                             float* __restrict__ den, const int* __restrict__ dst,
                             int pairs, int H) {
  int i = blockIdx.x * blockDim.x + threadIdx.x;
  if (i >= pairs) return;
  int e = i / H, h = i - e * H;
  int t = dst[e];
  float ex = expf(sc[i] - m[(size_t)t * H + h]);
  sc[i] = ex;
  atomicAdd(den + (size_t)t * H + h, ex);
}

// ---------------------------------------------------------------- out[dst] += (ex/den) * v[src]
__global__ void edge_aggregate(const float* __restrict__ sc, const float* __restrict__ den,
                               const float* __restrict__ v,
                               const int* __restrict__ src, const int* __restrict__ dst,
                               float* __restrict__ out, long total, int H, int d) {
  long i = (long)blockIdx.x * blockDim.x + threadIdx.x;
  if (i >= total) return;
  int f = (int)(i % d);
  int p = (int)(i / d);
  int e = p / H, h = p - e * H;
  int s = src[e], t = dst[e];
  float alpha = sc[p] / (den[(size_t)t * H + h] + 1e-16f);
  atomicAdd(out + (size_t)t * H * d + h * d + f, alpha * v[(size_t)s * H * d + h * d + f]);
}

// ---------------------------------------------------------------- relu in place
__global__ void relu_k(float* __restrict__ x, long n) {
  long i = (long)blockIdx.x * blockDim.x + threadIdx.x;
  if (i < n) x[i] = fmaxf(x[i], 0.0f);
}

// ================================================================ driver
extern "C" void kernel_launch(void* const* d_in, const int* in_sizes, int n_in,
                              void* d_out, int out_size, void* d_ws, size_t ws_size,
                              hipStream_t stream) {
  (void)in_sizes; (void)n_in; (void)out_size; (void)ws_size;

  const float* x  = (const float*)d_in[0];
  const float* pe = (const float*)d_in[1];
  const int*   ei = (const int*)d_in[2];
  const int* src = ei;
  const int* dst = ei + EE;

  // weights: per layer l, order q,k,v,s with (w,b) pairs
  const float* Wm[3][4];
  const float* Bm[3][4];
  int idx = 3;
  for (int l = 0; l < 3; ++l)
    for (int j = 0; j < 4; ++j) {
      Wm[l][j] = (const float*)d_in[idx++];
      Bm[l][j] = (const float*)d_in[idx++];
    }

  float* ws  = (float*)d_ws;
  float* h0  = ws;  ws += (size_t)NN * 136;   // concat input, stride 136
  float* hA  = ws;  ws += (size_t)NN * 256;   // layer1 output
  float* hB  = ws;  ws += (size_t)NN * 256;   // layer2 output
  float* q   = ws;  ws += (size_t)NN * 256;
  float* k   = ws;  ws += (size_t)NN * 256;
  float* v   = ws;  ws += (size_t)NN * 256;
  float* sc  = ws;  ws += (size_t)EE * 4;     // scores / exp, E*H
  float* m   = ws;  ws += (size_t)NN * 4;     // segment max, N*H
  float* den = ws;                            // segment sum, N*H

  // build [x | pe]
  {
    long tot = (long)NN * 136;
    concat_kernel<<<dim3((unsigned)((tot + 255) / 256)), dim3(256), 0, stream>>>(x, pe, h0, NN);
  }

  auto run_layer = [&](const float* in, int fin, int H, int d,
                       const float* const* Wl, const float* const* Bl,
                       float* out, bool do_relu) {
    const int fout = H * d;
    dim3 gblk(32, fout / 64);
    dim3 ggrd(NN / 16);
    // q / k / v projections
    wmma_gemm_bias<<<ggrd, gblk, 0, stream>>>(in, fin, Wl[0], Bl[0], q,   fout, fin);
    wmma_gemm_bias<<<ggrd, gblk, 0, stream>>>(in, fin, Wl[1], Bl[1], k,   fout, fin);
    wmma_gemm_bias<<<ggrd, gblk, 0, stream>>>(in, fin, Wl[2], Bl[2], v,   fout, fin);
    // skip projection straight into the layer output; edge aggregation adds on top
    wmma_gemm_bias<<<ggrd, gblk, 0, stream>>>(in, fin, Wl[3], Bl[3], out, fout, fin);

    const int NH = NN * H;
    init_m_den<<<dim3((NH + 255) / 256), dim3(256), 0, stream>>>(m, den, NH);

    const int pairs = EE * H;
    const float scale = 1.0f / sqrtf((float)d);
    edge_score_max<<<dim3((pairs + 7) / 8), dim3(256), 0, stream>>>(q, k, src, dst, sc, m,
                                                                    pairs, H, d, scale);
    edge_exp_sum<<<dim3((pairs + 255) / 256), dim3(256), 0, stream>>>(sc, m, den, dst, pairs, H);

    const long tot = (long)pairs * d;
    edge_aggregate<<<dim3((unsigned)((tot + 255) / 256)), dim3(256), 0, stream>>>(
        sc, den, v, src, dst, out, tot, H, d);

    if (do_relu) {
      long nel = (long)NN * fout;
      relu_k<<<dim3((unsigned)((nel + 255) / 256)), dim3(256), 0, stream>>>(out, nel);
    }
  };

  run_layer(h0, 136, 4, 64,  Wm[0], Bm[0], hA, true);
  run_layer(hA, 256, 4, 64,  Wm[1], Bm[1], hB, true);
  run_layer(hB, 256, 1, 128, Wm[2], Bm[2], (float*)d_out, false);
}